// GraphAttentionConvLayer_42314017800357
// MI455X (gfx1250) — compile-verified
//
#include <hip/hip_runtime.h>

// ---------------- problem constants (match reference) ----------------
#define BATCH   8
#define NPTS    4096
#define NPOINT  1024
#define NSAMPLE 32
#define FEATC   64
#define PTCH    67          // 3 xyz + 64 feat
#define C2      128         // final MLP width
#define OUTC    131         // 3 + 128
#define LSTR    132         // LDS row stride (padded, avoids bank conflicts)
#define RAD2    0.04f       // 0.2^2
#define BIGF    1.0e10f
#define SENT    3.0e10f
#define BNSCL   0.99999500003749965f   // 1/sqrt(1 + 1e-5)
#define LALPHA  0.2f

typedef float v2f __attribute__((ext_vector_type(2)));
typedef float v8f __attribute__((ext_vector_type(8)));

// =====================================================================
// Kernel 1: farthest point sampling. One workgroup (256 thr) per batch.
// xyz in LDS, per-thread distance registers, LDS tree argmax per step.
// =====================================================================
__global__ void __launch_bounds__(256) fps_kernel(const float* __restrict__ pts,
                                                  int* __restrict__ cidx,
                                                  float* __restrict__ cxyz) {
  __shared__ float sx[NPTS], sy[NPTS], sz[NPTS];
  __shared__ float rv[256];
  __shared__ int   ri[256];
  const int b = blockIdx.x, t = threadIdx.x;
  const float* base = pts + (size_t)b * NPTS * PTCH;

  float dloc[NPTS / 256];
#pragma unroll
  for (int j = 0; j < NPTS / 256; ++j) dloc[j] = BIGF;

  for (int i = t; i < NPTS; i += 256) {
    sx[i] = base[i * PTCH + 0];
    sy[i] = base[i * PTCH + 1];
    sz[i] = base[i * PTCH + 2];
  }
  __syncthreads();

  int far = 0;
  for (int it = 0; it < NPOINT; ++it) {
    const float fx = sx[far], fy = sy[far], fz = sz[far];
    if (t == 0) {
      cidx[b * NPOINT + it] = far;
      cxyz[(b * NPOINT + it) * 3 + 0] = fx;
      cxyz[(b * NPOINT + it) * 3 + 1] = fy;
      cxyz[(b * NPOINT + it) * 3 + 2] = fz;
    }
    float best = -1.0f; int bi = 0;
#pragma unroll
    for (int j = 0; j < NPTS / 256; ++j) {
      const int i = t + j * 256;
      const float dx = sx[i] - fx, dy = sy[i] - fy, dz = sz[i] - fz;
      const float dd = dx * dx + dy * dy + dz * dz;
      const float nd = fminf(dloc[j], dd);
      dloc[j] = nd;
      if (nd > best) { best = nd; bi = i; }
    }
    rv[t] = best; ri[t] = bi;
    __syncthreads();
    for (int s = 128; s > 0; s >>= 1) {
      if (t < s && rv[t + s] > rv[t]) { rv[t] = rv[t + s]; ri[t] = ri[t + s]; }
      __syncthreads();
    }
    far = ri[0];
    __syncthreads();
  }
}

// =====================================================================
// Kernel 2: ball query / top-NSAMPLE nearest within radius.
// One workgroup (128 thr) per center; distances in LDS; 32 argmin rounds.
// =====================================================================
__global__ void __launch_bounds__(128) qball_kernel(const float* __restrict__ pts,
                                                    const float* __restrict__ cxyz,
                                                    int* __restrict__ gidx) {
  __shared__ float sd[NPTS];
  __shared__ float rv[128];
  __shared__ int   ri[128];
  const int c = blockIdx.x, t = threadIdx.x;
  const int b = c / NPOINT;
  const float px = cxyz[c * 3 + 0], py = cxyz[c * 3 + 1], pz = cxyz[c * 3 + 2];
  const float* base = pts + (size_t)b * NPTS * PTCH;

  for (int i = t; i < NPTS; i += 128) {
    const float dx = base[i * PTCH + 0] - px;
    const float dy = base[i * PTCH + 1] - py;
    const float dz = base[i * PTCH + 2] - pz;
    const float dd = dx * dx + dy * dy + dz * dz;
    sd[i] = (dd > RAD2) ? BIGF : dd;
  }
  __syncthreads();

  int first = 0;
  for (int k = 0; k < NSAMPLE; ++k) {
    float best = SENT; int bi = NPTS;
    for (int i = t; i < NPTS; i += 128) {
      const float v = sd[i];
      if (v < best || (v == best && i < bi)) { best = v; bi = i; }
    }
    rv[t] = best; ri[t] = bi;
    __syncthreads();
    for (int s = 64; s > 0; s >>= 1) {
      if (t < s) {
        if (rv[t + s] < rv[t] || (rv[t + s] == rv[t] && ri[t + s] < ri[t])) {
          rv[t] = rv[t + s]; ri[t] = ri[t + s];
        }
      }
      __syncthreads();
    }
    const int sel = ri[0];
    const float sval = rv[0];
    if (k == 0) first = sel;
    if (t == 0) gidx[c * NSAMPLE + k] = (sval >= 0.5f * BIGF) ? first : sel;
    __syncthreads();
    if (t == 0) sd[sel] = SENT;
    __syncthreads();
  }
}

// =====================================================================
// WMMA helper: one MLP layer on a 16-row tile starting at rowOff.
// X,Y are LDS tiles (row stride LSTR). W is (Cout,Cin) row-major.
// D = relu(((X * W^T) + b) * g/sqrt(1+eps) + be), via V_WMMA_F32_16X16X4_F32.
//  A layout: lane(l) holds row M=l%16; v2f = K {kb+2h, kb+2h+1}, h=l/16.
//  B layout: lane(l) holds col N=l%16; same K split.
//  C/D: VGPR r, lane l -> (M = r + 8h, N = l%16).
// =====================================================================
__device__ __forceinline__ void wmma_layer(const float* X, float* Y,
                                           const float* __restrict__ W,
                                           const float* __restrict__ bb,
                                           const float* __restrict__ gg,
                                           const float* __restrict__ be,
                                           int Cin, int Cout, int rowOff, int lane) {
  const int lp = lane & 15, half = lane >> 4;
  for (int nt = 0; nt < (Cout >> 4); ++nt) {
    v8f acc = {0.f, 0.f, 0.f, 0.f, 0.f, 0.f, 0.f, 0.f};
    const int ch = nt * 16 + lp;
    for (int kb = 0; kb < Cin; kb += 4) {
      const int ka = kb + 2 * half;
      v2f a, bv;
      a.x = X[(rowOff + lp) * LSTR + ka];
      a.y = X[(rowOff + lp) * LSTR + ka + 1];
      bv.x = W[ch * Cin + ka];
      bv.y = W[ch * Cin + ka + 1];
      acc = __builtin_amdgcn_wmma_f32_16x16x4_f32(false, a, false, bv,
                                                  (short)0, acc, false, false);
    }
    const float sb = bb[ch];
    const float s  = gg[ch] * BNSCL;
    const float sbe = be[ch];
#pragma unroll
    for (int r = 0; r < 8; ++r) {
      float v = (acc[r] + sb) * s + sbe;
      v = (v > 0.f) ? v : 0.f;
      Y[(rowOff + r + 8 * half) * LSTR + nt * 16 + lp] = v;
    }
  }
}

// =====================================================================
// Kernel 3: center-feature MLP. One wave per 16 centers.
// =====================================================================
__global__ void __launch_bounds__(32) cmlp_kernel(const float* __restrict__ pts,
                                                  const int* __restrict__ cidx,
                                                  const float* W0, const float* b0, const float* g0, const float* be0,
                                                  const float* W1, const float* b1, const float* g1, const float* be1,
                                                  const float* W2, const float* b2, const float* g2, const float* be2,
                                                  float* __restrict__ cmlp) {
  __shared__ float X0[16 * LSTR], X1[16 * LSTR];
  const int lane = threadIdx.x;
  const int cbase = blockIdx.x * 16;
  const int b = cbase / NPOINT;
  const float* pb = pts + (size_t)b * NPTS * PTCH;

  for (int idx = lane; idx < 16 * FEATC; idx += 32) {
    const int r = idx >> 6, f = idx & 63;
    const int gi = cidx[cbase + r];
    X0[r * LSTR + f] = pb[gi * PTCH + 3 + f];
  }
  __syncthreads();

  wmma_layer(X0, X1, W0, b0, g0, be0, 64, 64, 0, lane);  __syncthreads();
  wmma_layer(X1, X0, W1, b1, g1, be1, 64, 64, 0, lane);  __syncthreads();
  wmma_layer(X0, X1, W2, b2, g2, be2, 64, 128, 0, lane); __syncthreads();

  for (int idx = lane; idx < 16 * C2; idx += 32) {
    const int r = idx >> 7, f = idx & 127;
    cmlp[(size_t)(cbase + r) * C2 + f] = X1[r * LSTR + f];
  }
}

// =====================================================================
// Kernel 4: fused group pipeline. One wave (32 lanes) per center.
// gather -> 3-layer WMMA MLP (2 M-tiles) -> delta -> WMMA attention GEMM
// -> leaky relu -> softmax over samples -> weighted sum -> output.
// LDS aliasing: delta reuses X0 (dead after MLP).
// =====================================================================
__global__ void __launch_bounds__(32) group_kernel(const float* __restrict__ pts,
                                                   const int* __restrict__ gidx,
                                                   const float* __restrict__ cxyz,
                                                   const float* __restrict__ cmlp,
                                                   const float* W0, const float* b0, const float* g0, const float* be0,
                                                   const float* W1, const float* b1, const float* g1, const float* be1,
                                                   const float* W2, const float* b2, const float* g2, const float* be2,
                                                   const float* __restrict__ a_att,
                                                   float* __restrict__ out) {
  __shared__ float gx[NSAMPLE * 3];
  __shared__ float X0[NSAMPLE * LSTR];   // input feats -> layer1 out -> delta
  __shared__ float X1[NSAMPLE * LSTR];   // layer0 out  -> final gfeat (32x128)
  __shared__ float ATT[NSAMPLE * LSTR];
  __shared__ float cpt[LSTR];

  const int lane = threadIdx.x;
  const int c = blockIdx.x;
  const int b = c / NPOINT;
  const float* pb = pts + (size_t)b * NPTS * PTCH;

  // gather neighbor xyz + feats (one sample per lane)
  {
    const int gi = gidx[c * NSAMPLE + lane];
    gx[lane * 3 + 0] = pb[gi * PTCH + 0];
    gx[lane * 3 + 1] = pb[gi * PTCH + 1];
    gx[lane * 3 + 2] = pb[gi * PTCH + 2];
    for (int f = 0; f < FEATC; ++f) X0[lane * LSTR + f] = pb[gi * PTCH + 3 + f];
  }
  for (int i = lane; i < LSTR; i += 32)
    cpt[i] = (i < 3) ? cxyz[c * 3 + i]
                     : ((i < OUTC) ? cmlp[(size_t)c * C2 + (i - 3)] : 0.0f);
  __syncthreads();

  // 3-layer MLP on both 16-row tiles (rows 16-31 of X0 stay intact for mt=1)
  for (int mt = 0; mt < 2; ++mt) {
    const int ro = mt * 16;
    wmma_layer(X0, X1, W0, b0, g0, be0, 64, 64, ro, lane);  __syncthreads();
    wmma_layer(X1, X0, W1, b1, g1, be1, 64, 64, ro, lane);  __syncthreads();
    wmma_layer(X0, X1, W2, b2, g2, be2, 64, 128, ro, lane); __syncthreads();
  }
  // X1 now holds final gfeat (32 x 128)

  // delta = [gxyz, gfeat] - cpts  (reuse X0; pad K col 131 with zero)
  {
    const int s = lane;
    X0[s * LSTR + 0] = gx[s * 3 + 0] - cpt[0];
    X0[s * LSTR + 1] = gx[s * 3 + 1] - cpt[1];
    X0[s * LSTR + 2] = gx[s * 3 + 2] - cpt[2];
    for (int d = 0; d < C2; ++d)
      X0[s * LSTR + 3 + d] = X1[s * LSTR + d] - cpt[3 + d];
    X0[s * LSTR + 131] = 0.0f;
  }
  __syncthreads();

  // attention GEMM: ATT = leaky_relu( delta(32x132) @ a_att_pad(132x128) )
  const int lp = lane & 15, half = lane >> 4;
  for (int mt = 0; mt < 2; ++mt) {
    const int ro = mt * 16;
    for (int nt = 0; nt < 8; ++nt) {
      v8f acc = {0.f, 0.f, 0.f, 0.f, 0.f, 0.f, 0.f, 0.f};
      const int ch = nt * 16 + lp;
      for (int kb = 0; kb < LSTR; kb += 4) {
        const int ka = kb + 2 * half;
        v2f a, bv;
        a.x = X0[(ro + lp) * LSTR + ka];
        a.y = X0[(ro + lp) * LSTR + ka + 1];
        const int k0 = (ka     < OUTC) ? ka       : (OUTC - 1);
        const int k1 = (ka + 1 < OUTC) ? (ka + 1) : (OUTC - 1);
        bv.x = a_att[k0 * C2 + ch] * ((ka     < OUTC) ? 1.0f : 0.0f);
        bv.y = a_att[k1 * C2 + ch] * ((ka + 1 < OUTC) ? 1.0f : 0.0f);
        acc = __builtin_amdgcn_wmma_f32_16x16x4_f32(false, a, false, bv,
                                                    (short)0, acc, false, false);
      }
#pragma unroll
      for (int r = 0; r < 8; ++r) {
        float v = acc[r];
        v = (v > 0.0f) ? v : (LALPHA * v);
        ATT[(ro + r + 8 * half) * LSTR + nt * 16 + lp] = v;
      }
    }
  }
  __syncthreads();

  // softmax over the 32 samples per channel + weighted sum with gfeat
#pragma unroll
  for (int dd = 0; dd < 4; ++dd) {
    const int d = lane + dd * 32;
    float m = -3.4e38f;
    for (int s = 0; s < NSAMPLE; ++s) m = fmaxf(m, ATT[s * LSTR + d]);
    float Z = 0.0f, acc = 0.0f;
    for (int s = 0; s < NSAMPLE; ++s) {
      const float e = __expf(ATT[s * LSTR + d] - m);
      Z += e;
      acc += e * X1[s * LSTR + d];
    }
    out[(size_t)c * OUTC + 3 + d] = acc / Z;
  }
  if (lane < 3) out[(size_t)c * OUTC + lane] = cpt[lane];
}

// =====================================================================
// host launcher
// =====================================================================
extern "C" void kernel_launch(void* const* d_in, const int* in_sizes, int n_in,
                              void* d_out, int out_size, void* d_ws, size_t ws_size,
                              hipStream_t stream) {
  (void)in_sizes; (void)n_in; (void)out_size; (void)ws_size;

  const float* pts   = (const float*)d_in[0];
  const float* W0    = (const float*)d_in[1];
  const float* b0    = (const float*)d_in[2];
  const float* g0    = (const float*)d_in[3];
  const float* be0   = (const float*)d_in[4];
  const float* W1    = (const float*)d_in[5];
  const float* b1    = (const float*)d_in[6];
  const float* g1    = (const float*)d_in[7];
  const float* be1   = (const float*)d_in[8];
  const float* W2    = (const float*)d_in[9];
  const float* b2    = (const float*)d_in[10];
  const float* g2    = (const float*)d_in[11];
  const float* be2   = (const float*)d_in[12];
  const float* a_att = (const float*)d_in[13];
  float* out = (float*)d_out;

  // workspace layout (256B-aligned slices)
  char* ws = (char*)d_ws;
  size_t off = 0;
  int* cidx = (int*)(ws + off);   off += (size_t)BATCH * NPOINT * sizeof(int);            off = (off + 255) & ~(size_t)255;
  int* gidx = (int*)(ws + off);   off += (size_t)BATCH * NPOINT * NSAMPLE * sizeof(int);  off = (off + 255) & ~(size_t)255;
  float* cxyz = (float*)(ws + off); off += (size_t)BATCH * NPOINT * 3 * sizeof(float);    off = (off + 255) & ~(size_t)255;
  float* cmlp = (float*)(ws + off);

  fps_kernel<<<BATCH, 256, 0, stream>>>(pts, cidx, cxyz);
  qball_kernel<<<BATCH * NPOINT, 128, 0, stream>>>(pts, cxyz, gidx);
  cmlp_kernel<<<(BATCH * NPOINT) / 16, 32, 0, stream>>>(pts, cidx,
      W0, b0, g0, be0, W1, b1, g1, be1, W2, b2, g2, be2, cmlp);
  group_kernel<<<BATCH * NPOINT, 32, 0, stream>>>(pts, gidx, cxyz, cmlp,
      W0, b0, g0, be0, W1, b1, g1, be1, W2, b2, g2, be2, a_att, out);
}